// GATModel_18090402251155
// MI455X (gfx1250) — compile-verified
//
#include <hip/hip_runtime.h>

#define N_NODES 100000
#define E_EDGES 1600000
#define ET (E_EDGES + N_NODES)   // edges + self loops
#define IN_C 128
#define HID 32
#define HEADS 4
#define H1C 128                   // HEADS*HID
#define OUT_C 16
#define NEG_SLOPE 0.2f

typedef __attribute__((ext_vector_type(2))) float v2f;
typedef __attribute__((ext_vector_type(8))) float v8f;
typedef __attribute__((ext_vector_type(4))) int   v4i;

#if __has_builtin(__builtin_amdgcn_global_load_async_to_lds_b128) && \
    __has_builtin(__builtin_amdgcn_s_wait_asynccnt)
#define HAVE_ASYNC_LDS 1
// builtin signature (from hipcc diagnostic): (v4i AS1*, v4i AS3*, imm off, imm cpol)
#define GPTR128(p) ((__attribute__((address_space(1))) v4i*)(p))
#define LPTR128(p) ((__attribute__((address_space(3))) v4i*)(p))
#else
#define HAVE_ASYNC_LDS 0
#endif

// ---- order-preserving float<->uint encoding for atomic max ----
static __device__ __forceinline__ unsigned encf(float x) {
    unsigned u = __float_as_uint(x);
    return (u & 0x80000000u) ? ~u : (u | 0x80000000u);
}
static __device__ __forceinline__ float decf(unsigned u) {
    return (u & 0x80000000u) ? __uint_as_float(u & 0x7fffffffu)
                             : __uint_as_float(~u);
}

static __device__ __forceinline__ void edge_sd(const long long* __restrict__ ei,
                                               long long e, int& s, int& d) {
    if (e < E_EDGES) { s = (int)ei[e]; d = (int)ei[E_EDGES + e]; }
    else             { s = d = (int)(e - E_EDGES); }           // self-loop
}

// ---------------- zero-init three spans (ws accumulators + d_out) ------------
__global__ void zero3_kernel(float* p0, size_t n0, float* p1, size_t n1,
                             float* p2, size_t n2) {
    size_t stride = (size_t)gridDim.x * blockDim.x;
    size_t i0 = (size_t)blockIdx.x * blockDim.x + threadIdx.x;
    for (size_t i = i0; i < n0; i += stride) p0[i] = 0.0f;
    for (size_t i = i0; i < n1; i += stride) p1[i] = 0.0f;
    for (size_t i = i0; i < n2; i += stride) p2[i] = 0.0f;
}

// ---------------- GEMM1: h1[N,128] = x[N,128] @ W1[128,128] (WMMA f32) -------
// Block = 8 waves; wave w owns rows [blk*128 + w*16, +16), all 8 col tiles.
// W1 staged in LDS in two 64-row chunks. LDS layout is K-PAIRED:
//   lWp[(k>>1)*256 + col*2 + (k&1)] = W[kc+k][col]
// so each lane's B fragment (rows k,k+1 @ col) is a single aligned ds_load_b64.
__global__ __launch_bounds__(256)
void gemm1_kernel(const float* __restrict__ x, const float* __restrict__ W,
                  float* __restrict__ h) {
    __shared__ float lWp[64 * H1C];     // 32KB paired layout
#if HAVE_ASYNC_LDS
    __shared__ float stage[64 * H1C];   // 32KB async landing buffer
#endif
    const int lane = threadIdx.x & 31;
    const int wave = threadIdx.x >> 5;
    const int half = lane >> 4;
    const int m    = lane & 15;
    const int row0 = (blockIdx.x * 8 + wave) * 16;
    const bool active = row0 < N_NODES;   // wave-uniform -> EXEC all ones inside

    v8f acc[8] = {};

    for (int kc = 0; kc < IN_C; kc += 64) {
        __syncthreads();
#if HAVE_ASYNC_LDS
        {   // async DMA global -> LDS stage (row-major), then repack to pairs
            const float* src = W + (size_t)kc * H1C;
            for (int i = threadIdx.x; i < 64 * H1C / 4; i += 256)
                __builtin_amdgcn_global_load_async_to_lds_b128(
                    GPTR128(src + i * 4), LPTR128(stage + i * 4), 0, 0);
            __builtin_amdgcn_s_wait_asynccnt(0);
            __syncthreads();
            for (int i = threadIdx.x; i < 64 * H1C / 4; i += 256) {
                int k = i >> 5, c = (i & 31) * 4;
                float4 v = *(const float4*)(stage + (size_t)k * H1C + c);
                float* dst = lWp + (size_t)(k >> 1) * (2 * H1C) + (k & 1);
                dst[(c + 0) * 2] = v.x; dst[(c + 1) * 2] = v.y;
                dst[(c + 2) * 2] = v.z; dst[(c + 3) * 2] = v.w;
            }
        }
#else
        for (int i = threadIdx.x; i < 64 * H1C / 4; i += 256) {
            int k = i >> 5, c = (i & 31) * 4;
            float4 v = *(const float4*)(W + (size_t)(kc + k) * H1C + c);
            float* dst = lWp + (size_t)(k >> 1) * (2 * H1C) + (k & 1);
            dst[(c + 0) * 2] = v.x; dst[(c + 1) * 2] = v.y;
            dst[(c + 2) * 2] = v.z; dst[(c + 3) * 2] = v.w;
        }
#endif
        __syncthreads();
        if (active) {
            const float* xrow = x + (size_t)(row0 + m) * IN_C;
            __builtin_prefetch(xrow + kc, 0, 0);
            for (int kk = 0; kk < 16; ++kk) {
                const int k0 = kc + kk * 4;
                float2 av = *(const float2*)(xrow + k0 + 2 * half);
                v2f a; a.x = av.x; a.y = av.y;
                const int p = kk * 2 + half;            // K-pair index
                const float* bp = lWp + (size_t)p * (2 * H1C) + m * 2;
                for (int t = 0; t < 8; ++t) {
                    v2f b = *(const v2f*)(bp + t * 32);  // ds_load_b64
                    acc[t] = __builtin_amdgcn_wmma_f32_16x16x4_f32(
                        false, a, false, b, (short)0, acc[t], false, false);
                }
            }
        }
    }
    if (active) {
        for (int t = 0; t < 8; ++t)
            for (int v = 0; v < 8; ++v)
                h[(size_t)(row0 + v + 8 * half) * H1C + t * 16 + m] = acc[t][v];
    }
}

// ------- GEMM2: h2[N,16] = relu(o1[N,128] + b1) @ W2[128,16] (WMMA f32) -----
__global__ __launch_bounds__(256)
void gemm2_kernel(const float* __restrict__ o1, const float* __restrict__ b1,
                  const float* __restrict__ W2, float* __restrict__ h2) {
    __shared__ float lWp[H1C * OUT_C];  // 8KB, K-paired layout
    const int lane = threadIdx.x & 31;
    const int wave = threadIdx.x >> 5;
    const int half = lane >> 4;
    const int m    = lane & 15;
    const int row0 = (blockIdx.x * 8 + wave) * 16;
    const bool active = row0 < N_NODES;

    for (int i = threadIdx.x; i < H1C * OUT_C / 4; i += 256) {
        int k = i >> 2, c = (i & 3) * 4;
        float4 v = ((const float4*)W2)[i];
        float* dst = lWp + (size_t)(k >> 1) * (2 * OUT_C) + (k & 1);
        dst[(c + 0) * 2] = v.x; dst[(c + 1) * 2] = v.y;
        dst[(c + 2) * 2] = v.z; dst[(c + 3) * 2] = v.w;
    }
    __syncthreads();

    if (active) {
        v8f acc = {};
        const float* orow = o1 + (size_t)(row0 + m) * H1C;
        for (int kk = 0; kk < 32; ++kk) {
            const int p  = kk * 2 + half;               // K-pair index
            const int k0 = 2 * p;
            float2 av = *(const float2*)(orow + k0);
            float2 bv = *(const float2*)(b1 + k0);
            v2f a;
            a.x = fmaxf(av.x + bv.x, 0.0f);             // +bias then ReLU, fused
            a.y = fmaxf(av.y + bv.y, 0.0f);
            v2f b = *(const v2f*)(lWp + (size_t)p * (2 * OUT_C) + m * 2);
            acc = __builtin_amdgcn_wmma_f32_16x16x4_f32(
                false, a, false, b, (short)0, acc, false, false);
        }
        for (int v = 0; v < 8; ++v)
            h2[(size_t)(row0 + v + 8 * half) * OUT_C + m] = acc[v];
    }
}

// ------------- attention logits: als/ald[n,h] = <h1[n,h,:], a_src/dst[h,:]> --
__global__ void alpha1_kernel(const float* __restrict__ h1,
                              const float* __restrict__ a_src,
                              const float* __restrict__ a_dst,
                              float* __restrict__ als, float* __restrict__ ald) {
    int t = blockIdx.x * blockDim.x + threadIdx.x;
    if (t >= N_NODES * HEADS) return;
    int n = t >> 2, hh = t & 3;
    const float* hp = h1 + (size_t)n * H1C + hh * HID;
    const float* sp = a_src + hh * HID;
    const float* dp = a_dst + hh * HID;
    float s = 0.f, d = 0.f;
    for (int c = 0; c < HID; c += 4) {
        float4 hv = *(const float4*)(hp + c);
        float4 sv = *(const float4*)(sp + c);
        float4 dv = *(const float4*)(dp + c);
        s += hv.x * sv.x + hv.y * sv.y + hv.z * sv.z + hv.w * sv.w;
        d += hv.x * dv.x + hv.y * dv.y + hv.z * dv.z + hv.w * dv.w;
    }
    als[t] = s; ald[t] = d;
}

__global__ void alpha2_kernel(const float* __restrict__ h2,
                              const float* __restrict__ a_src,
                              const float* __restrict__ a_dst,
                              float* __restrict__ als, float* __restrict__ ald) {
    int n = blockIdx.x * blockDim.x + threadIdx.x;
    if (n >= N_NODES) return;
    const float* hp = h2 + (size_t)n * OUT_C;
    float s = 0.f, d = 0.f;
    for (int c = 0; c < OUT_C; c += 4) {
        float4 hv = *(const float4*)(hp + c);
        float4 sv = *(const float4*)(a_src + c);
        float4 dv = *(const float4*)(a_dst + c);
        s += hv.x * sv.x + hv.y * sv.y + hv.z * sv.z + hv.w * sv.w;
        d += hv.x * dv.x + hv.y * dv.y + hv.z * dv.z + hv.w * dv.w;
    }
    als[n] = s; ald[n] = d;
}

// ------------------------- layer-1 edge passes -------------------------------
__global__ void edgemax1_kernel(const long long* __restrict__ ei,
                                const float* __restrict__ als,
                                const float* __restrict__ ald,
                                unsigned* __restrict__ m1) {
    int t = blockIdx.x * blockDim.x + threadIdx.x;
    if (t >= ET * HEADS) return;
    long long e = (long long)(t >> 2); int hh = t & 3;
    int s, d; edge_sd(ei, e, s, d);
    float sc = als[s * 4 + hh] + ald[d * 4 + hh];
    sc = sc > 0.f ? sc : NEG_SLOPE * sc;
    atomicMax(&m1[d * 4 + hh], encf(sc));
}

__global__ void edgesum1_kernel(const long long* __restrict__ ei,
                                const float* __restrict__ als,
                                const float* __restrict__ ald,
                                const unsigned* __restrict__ m1,
                                float* __restrict__ d1) {
    int t = blockIdx.x * blockDim.x + threadIdx.x;
    if (t >= ET * HEADS) return;
    long long e = (long long)(t >> 2); int hh = t & 3;
    int s, d; edge_sd(ei, e, s, d);
    float sc = als[s * 4 + hh] + ald[d * 4 + hh];
    sc = sc > 0.f ? sc : NEG_SLOPE * sc;
    atomicAdd(&d1[d * 4 + hh], expf(sc - decf(m1[d * 4 + hh])));
}

// one wave per edge; lane owns 4 consecutive channels (head = lane/8)
__global__ void edgeaggr1_kernel(const long long* __restrict__ ei,
                                 const float* __restrict__ als,
                                 const float* __restrict__ ald,
                                 const unsigned* __restrict__ m1,
                                 const float* __restrict__ d1,
                                 const float* __restrict__ h1,
                                 float* __restrict__ o1) {
    int t = blockIdx.x * blockDim.x + threadIdx.x;
    long long e = (long long)(t >> 5);
    if (e >= ET) return;
    int lane = t & 31;
    int s, d; edge_sd(ei, e, s, d);
    int hh = lane >> 3;
    float sc = als[s * 4 + hh] + ald[d * 4 + hh];
    sc = sc > 0.f ? sc : NEG_SLOPE * sc;
    float alpha = expf(sc - decf(m1[d * 4 + hh])) / d1[d * 4 + hh];
    int c0 = lane * 4;
    float4 hv = *(const float4*)(h1 + (size_t)s * H1C + c0);
    float* op = o1 + (size_t)d * H1C + c0;
    atomicAdd(op + 0, hv.x * alpha);
    atomicAdd(op + 1, hv.y * alpha);
    atomicAdd(op + 2, hv.z * alpha);
    atomicAdd(op + 3, hv.w * alpha);
}

// ------------------------- layer-2 edge passes (1 head, 16ch) ----------------
__global__ void edgemax2_kernel(const long long* __restrict__ ei,
                                const float* __restrict__ als,
                                const float* __restrict__ ald,
                                unsigned* __restrict__ m2) {
    int t = blockIdx.x * blockDim.x + threadIdx.x;
    if (t >= ET) return;
    int s, d; edge_sd(ei, (long long)t, s, d);
    float sc = als[s] + ald[d];
    sc = sc > 0.f ? sc : NEG_SLOPE * sc;
    atomicMax(&m2[d], encf(sc));
}

__global__ void edgesum2_kernel(const long long* __restrict__ ei,
                                const float* __restrict__ als,
                                const float* __restrict__ ald,
                                const unsigned* __restrict__ m2,
                                float* __restrict__ d2) {
    int t = blockIdx.x * blockDim.x + threadIdx.x;
    if (t >= ET) return;
    int s, d; edge_sd(ei, (long long)t, s, d);
    float sc = als[s] + ald[d];
    sc = sc > 0.f ? sc : NEG_SLOPE * sc;
    atomicAdd(&d2[d], expf(sc - decf(m2[d])));
}

// 4 threads per edge, each owns 4 channels
__global__ void edgeaggr2_kernel(const long long* __restrict__ ei,
                                 const float* __restrict__ als,
                                 const float* __restrict__ ald,
                                 const unsigned* __restrict__ m2,
                                 const float* __restrict__ d2,
                                 const float* __restrict__ h2,
                                 float* __restrict__ out) {
    int t = blockIdx.x * blockDim.x + threadIdx.x;
    long long e = (long long)(t >> 2);
    if (e >= ET) return;
    int c0 = (t & 3) * 4;
    int s, d; edge_sd(ei, e, s, d);
    float sc = als[s] + ald[d];
    sc = sc > 0.f ? sc : NEG_SLOPE * sc;
    float alpha = expf(sc - decf(m2[d])) / d2[d];
    float4 hv = *(const float4*)(h2 + (size_t)s * OUT_C + c0);
    float* op = out + (size_t)d * OUT_C + c0;
    atomicAdd(op + 0, hv.x * alpha);
    atomicAdd(op + 1, hv.y * alpha);
    atomicAdd(op + 2, hv.z * alpha);
    atomicAdd(op + 3, hv.w * alpha);
}

__global__ void bias2_kernel(float* __restrict__ out, const float* __restrict__ b2) {
    int i = blockIdx.x * blockDim.x + threadIdx.x;
    if (i < N_NODES * OUT_C) out[i] += b2[i & 15];
}

// -----------------------------------------------------------------------------
extern "C" void kernel_launch(void* const* d_in, const int* in_sizes, int n_in,
                              void* d_out, int out_size, void* d_ws, size_t ws_size,
                              hipStream_t stream) {
    const float*     x      = (const float*)d_in[0];
    const long long* ei     = (const long long*)d_in[1];   // int64 [2,E]
    const float*     W1     = (const float*)d_in[2];
    const float*     a_src1 = (const float*)d_in[3];
    const float*     a_dst1 = (const float*)d_in[4];
    const float*     b1     = (const float*)d_in[5];
    const float*     W2     = (const float*)d_in[6];
    const float*     a_src2 = (const float*)d_in[7];
    const float*     a_dst2 = (const float*)d_in[8];
    const float*     b2     = (const float*)d_in[9];
    float* out = (float*)d_out;

    // workspace carve-up (floats)
    float* ws = (float*)d_ws;
    float*    h1   = ws;                 ws += (size_t)N_NODES * H1C;
    float*    als1 = ws;                 ws += (size_t)N_NODES * HEADS;
    float*    ald1 = ws;                 ws += (size_t)N_NODES * HEADS;
    unsigned* m1   = (unsigned*)ws;      ws += (size_t)N_NODES * HEADS;
    float*    d1   = ws;                 ws += (size_t)N_NODES * HEADS;
    float*    o1   = ws;                 ws += (size_t)N_NODES * H1C;
    float*    h2   = ws;                 ws += (size_t)N_NODES * OUT_C;
    float*    als2 = ws;                 ws += (size_t)N_NODES;
    float*    ald2 = ws;                 ws += (size_t)N_NODES;
    unsigned* m2   = (unsigned*)ws;      ws += (size_t)N_NODES;
    float*    d2   = ws;                 ws += (size_t)N_NODES;

    const int B = 256;
    // zero: [m1,d1,o1] contiguous span, [m2,d2] contiguous span, d_out
    zero3_kernel<<<2048, B, 0, stream>>>(
        (float*)m1, (size_t)N_NODES * (HEADS * 2 + H1C),
        (float*)m2, (size_t)N_NODES * 2,
        out,        (size_t)N_NODES * OUT_C);

    const int gemmBlocks = (N_NODES / 16 + 7) / 8;   // 6250 waves -> 782 blocks
    gemm1_kernel<<<gemmBlocks, B, 0, stream>>>(x, W1, h1);
    alpha1_kernel<<<(N_NODES * HEADS + B - 1) / B, B, 0, stream>>>(h1, a_src1, a_dst1, als1, ald1);

    const int et4 = ET * HEADS;
    edgemax1_kernel<<<(et4 + B - 1) / B, B, 0, stream>>>(ei, als1, ald1, m1);
    edgesum1_kernel<<<(et4 + B - 1) / B, B, 0, stream>>>(ei, als1, ald1, m1, d1);
    edgeaggr1_kernel<<<ET / 8, B, 0, stream>>>(ei, als1, ald1, m1, d1, h1, o1); // ET*32/256

    gemm2_kernel<<<gemmBlocks, B, 0, stream>>>(o1, b1, W2, h2);
    alpha2_kernel<<<(N_NODES + B - 1) / B, B, 0, stream>>>(h2, a_src2, a_dst2, als2, ald2);

    edgemax2_kernel<<<(ET + B - 1) / B, B, 0, stream>>>(ei, als2, ald2, m2);
    edgesum2_kernel<<<(ET + B - 1) / B, B, 0, stream>>>(ei, als2, ald2, m2, d2);
    edgeaggr2_kernel<<<(ET * 4 + B - 1) / B, B, 0, stream>>>(ei, als2, ald2, m2, d2, h2, out);

    bias2_kernel<<<(N_NODES * OUT_C + B - 1) / B, B, 0, stream>>>(out, b2);
    (void)in_sizes; (void)n_in; (void)out_size; (void)ws_size;
}